// Model_32847909880089
// MI455X (gfx1250) — compile-verified
//
#include <hip/hip_runtime.h>
#include <math.h>

typedef __attribute__((ext_vector_type(2))) float v2f;
typedef __attribute__((ext_vector_type(8))) float v8f;

// ---------------------------------------------------------------------------
// init: deg = 1.0 (self loop), pool sums/counts = 0
// ---------------------------------------------------------------------------
__global__ void k_init(float* __restrict__ deg, float* __restrict__ poolS,
                       float* __restrict__ poolC, int N, int G) {
    int t = blockIdx.x * blockDim.x + threadIdx.x;
    if (t < N) deg[t] = 1.0f;
    int u = t - N;
    if (u >= 0 && u < G * 16) poolS[u] = 0.0f;
    int w = u - G * 16;
    if (w >= 0 && w < G) poolC[w] = 0.0f;
}

// ---------------------------------------------------------------------------
// degree accumulation over edges (dst side)
// ---------------------------------------------------------------------------
__global__ void k_degree(const int* __restrict__ dst, float* __restrict__ deg, int E) {
    int e = blockIdx.x * blockDim.x + threadIdx.x;
    if (e < E) atomicAdd(&deg[dst[e]], 1.0f);
}

// deg -> rsqrt(deg)   (deg >= 1 always, self loops included)
__global__ void k_rsqrt(float* __restrict__ deg, int N) {
    int t = blockIdx.x * blockDim.x + threadIdx.x;
    if (t < N) deg[t] = rsqrtf(deg[t]);
}

// ---------------------------------------------------------------------------
// WMMA GEMM: hs[n][0:16] = dinv[n] * (in[n][0:KIN] @ W[0:16][0:KIN]^T)
// also writes acc = hs (scatter accumulator init == self-loop contribution).
// One wave32 per 16-row tile. V_WMMA_F32_16X16X4_F32, K stepped by 4.
// NOTE: `in` and `acc` may alias (layer 2): every store's data depends on all
// loads through the WMMA chain, and each wave touches only its own rows.
// ---------------------------------------------------------------------------
template <int KIN>
__global__ void k_gemm_wmma(const float* in,              // [N, KIN]
                            const float* __restrict__ W,  // [16, KIN] row-major
                            const float* __restrict__ dinv,
                            float* __restrict__ hs,       // [N, 16]
                            float* acc,                   // [N, 16]
                            int N) {
    const int lane = threadIdx.x & 31;
    const int wave = blockIdx.x * (blockDim.x >> 5) + (threadIdx.x >> 5);
    const int row0 = wave << 4;
    if (row0 >= N) return;                 // wave-uniform exit; EXEC stays full

    const int m  = lane & 15;              // M for A frags / N-col for B frags
    const int kl = (lane >> 4) << 1;       // upper half-wave holds K+2,K+3
    int row = row0 + m;
    if (row >= N) row = N - 1;             // clamp (only matters on tail tile)

    v8f c = {0.f, 0.f, 0.f, 0.f, 0.f, 0.f, 0.f, 0.f};
#pragma unroll
    for (int k = 0; k < KIN / 4; ++k) {
        // A: 16x4 tile of `in`; lane m holds in[row][4k+kl .. 4k+kl+1]
        v2f a = *reinterpret_cast<const v2f*>(in + (size_t)row * KIN + k * 4 + kl);
        // B: 4x16 tile of W^T; lane n=m holds W[n][4k+kl .. 4k+kl+1]
        v2f b = *reinterpret_cast<const v2f*>(W + (size_t)m * KIN + k * 4 + kl);
        c = __builtin_amdgcn_wmma_f32_16x16x4_f32(
            false, a, false, b, (short)0, c, false, false);
    }

    // C/D layout: VGPR v -> (M = v + 8*(lane>=16), N = lane&15)
    const int col   = lane & 15;
    const int rbase = row0 + ((lane >> 4) << 3);

    if (row0 + 16 <= N) {
        // Full tile (every wave when N % 16 == 0): straight-line epilogue.
        // rbase is 8-aligned -> 16B-aligned b128 loads of dinv.
        const float4 d0 = *reinterpret_cast<const float4*>(dinv + rbase);
        const float4 d1 = *reinterpret_cast<const float4*>(dinv + rbase + 4);
        const float dv[8] = {d0.x, d0.y, d0.z, d0.w, d1.x, d1.y, d1.z, d1.w};
#pragma unroll
        for (int v = 0; v < 8; ++v) {
            const float val = c[v] * dv[v];
            const size_t off = (size_t)(rbase + v) * 16 + col;
            hs[off]  = val;
            acc[off] = val;
        }
    } else {
        // Tail tile (only possible when N % 16 != 0): guarded.
#pragma unroll
        for (int v = 0; v < 8; ++v) {
            const int r = rbase + v;
            if (r < N) {
                const float val = c[v] * dinv[r];
                const size_t off = (size_t)r * 16 + col;
                hs[off]  = val;
                acc[off] = val;
            }
        }
    }
}

// ---------------------------------------------------------------------------
// edge scatter: acc[dst] += hs[src]   (normalization already factored out)
// 4 threads per edge, float4 gather + 4 atomic f32 adds each.
// ---------------------------------------------------------------------------
__global__ void k_scatter(const int* __restrict__ src, const int* __restrict__ dst,
                          const float* __restrict__ hs, float* __restrict__ acc, int E) {
    int t = blockIdx.x * blockDim.x + threadIdx.x;
    int e = t >> 2;
    if (e >= E) return;
    int q = (t & 3) << 2;
    int s = src[e];
    int d = dst[e];
    const float4 v = *reinterpret_cast<const float4*>(hs + (size_t)s * 16 + q);
    float* p = acc + (size_t)d * 16 + q;
    atomicAdd(p + 0, v.x);
    atomicAdd(p + 1, v.y);
    atomicAdd(p + 2, v.z);
    atomicAdd(p + 3, v.w);
}

// conv1 epilogue: acc = relu(dinv[n]*acc + b1[f])
__global__ void k_fin_relu(float* __restrict__ acc, const float* __restrict__ dinv,
                           const float* __restrict__ bias, int N) {
    int t = blockIdx.x * blockDim.x + threadIdx.x;
    if (t >= N * 16) return;
    int n = t >> 4, f = t & 15;
    float v = dinv[n] * acc[t] + bias[f];
    acc[t] = v > 0.0f ? v : 0.0f;
}

// conv2 epilogue fused with mean-pool accumulation
__global__ void k_fin_pool(const float* __restrict__ acc, const float* __restrict__ dinv,
                           const float* __restrict__ bias, const int* __restrict__ batch,
                           float* __restrict__ poolS, float* __restrict__ poolC, int N) {
    int t = blockIdx.x * blockDim.x + threadIdx.x;
    if (t >= N * 16) return;
    int n = t >> 4, f = t & 15;
    float v = dinv[n] * acc[t] + bias[f];
    int g = batch[n];
    atomicAdd(&poolS[(size_t)g * 16 + f], v);
    if (f == 0) atomicAdd(&poolC[g], 1.0f);
}

// per-graph head: mean, 16->5 linear, softmax
__global__ void k_head(const float* __restrict__ poolS, const float* __restrict__ poolC,
                       const float* __restrict__ Wl, const float* __restrict__ bl,
                       float* __restrict__ out, int G) {
    int g = blockIdx.x * blockDim.x + threadIdx.x;
    if (g >= G) return;
    float inv = 1.0f / fmaxf(poolC[g], 1.0f);
    float lg[5];
    float mx = -3.4e38f;
#pragma unroll
    for (int j = 0; j < 5; ++j) {
        float s = bl[j];
#pragma unroll
        for (int f = 0; f < 16; ++f) s += poolS[(size_t)g * 16 + f] * inv * Wl[j * 16 + f];
        lg[j] = s;
        mx = fmaxf(mx, s);
    }
    float se = 0.0f;
#pragma unroll
    for (int j = 0; j < 5; ++j) { lg[j] = __expf(lg[j] - mx); se += lg[j]; }
    float r = 1.0f / se;
#pragma unroll
    for (int j = 0; j < 5; ++j) out[(size_t)g * 5 + j] = lg[j] * r;
}

// ---------------------------------------------------------------------------
extern "C" void kernel_launch(void* const* d_in, const int* in_sizes, int n_in,
                              void* d_out, int out_size, void* d_ws, size_t ws_size,
                              hipStream_t stream) {
    (void)n_in; (void)ws_size;
    const float* x     = (const float*)d_in[0];   // [N,32]
    const int*   ei    = (const int*)  d_in[1];   // [2,E]
    const int*   batch = (const int*)  d_in[2];   // [N]
    const float* W1    = (const float*)d_in[3];   // [16,32]
    const float* b1    = (const float*)d_in[4];   // [16]
    const float* W2    = (const float*)d_in[5];   // [16,16]
    const float* b2    = (const float*)d_in[6];   // [16]
    const float* Wl    = (const float*)d_in[7];   // [5,16]
    const float* bl    = (const float*)d_in[8];   // [5]
    float* out = (float*)d_out;

    const int N = in_sizes[0] / 32;
    const int E = in_sizes[1] / 2;
    const int G = out_size / 5;
    const int* src = ei;
    const int* dst = ei + E;

    float* ws    = (float*)d_ws;
    float* dinv  = ws;                                  // N
    float* bufA  = dinv + N;                            // N*16 (hs)
    float* bufB  = bufA + (size_t)N * 16;               // N*16 (accumulator)
    float* poolS = bufB + (size_t)N * 16;               // G*16
    float* poolC = poolS + (size_t)G * 16;              // G

    const int B = 256;
    auto cdiv = [](long a, long b) { return (int)((a + b - 1) / b); };

    // degrees -> dinv
    k_init  <<<cdiv((long)N + (long)G * 17, B), B, 0, stream>>>(dinv, poolS, poolC, N, G);
    k_degree<<<cdiv(E, B), B, 0, stream>>>(dst, dinv, E);
    k_rsqrt <<<cdiv(N, B), B, 0, stream>>>(dinv, N);

    const int tiles      = cdiv(N, 16);
    const int gemmBlocks = cdiv(tiles, B / 32);

    // conv1
    k_gemm_wmma<32><<<gemmBlocks, B, 0, stream>>>(x, W1, dinv, bufA, bufB, N);
    k_scatter <<<cdiv((long)E * 4, B), B, 0, stream>>>(src, dst, bufA, bufB, E);
    k_fin_relu<<<cdiv((long)N * 16, B), B, 0, stream>>>(bufB, dinv, b1, N);

    // conv2 (bufB is both GEMM input and accumulator target; safe, see kernel)
    k_gemm_wmma<16><<<gemmBlocks, B, 0, stream>>>(bufB, W2, dinv, bufA, bufB, N);
    k_scatter <<<cdiv((long)E * 4, B), B, 0, stream>>>(src, dst, bufA, bufB, E);
    k_fin_pool<<<cdiv((long)N * 16, B), B, 0, stream>>>(bufB, dinv, b2, batch, poolS, poolC, N);

    // head
    k_head<<<cdiv(G, B), B, 0, stream>>>(poolS, poolC, Wl, bl, out, G);
}